// GCNLayer_36275293782548
// MI455X (gfx1250) — compile-verified
//
#include <hip/hip_runtime.h>

typedef float v2f __attribute__((ext_vector_type(2)));
typedef float v8f __attribute__((ext_vector_type(8)));

#define N_NODES 50000
#define IN_F    256
#define OUT_F   96
#define N_EDGES 800000

// ---------------------------------------------------------------------------
// Kernel 1: zero the aggregation buffer (harness poisons d_ws with 0xAA).
// ---------------------------------------------------------------------------
__global__ void gcn_zero_kernel(float4* __restrict__ agg, int n4) {
    int i = blockIdx.x * blockDim.x + threadIdx.x;
    if (i < n4) agg[i] = make_float4(0.f, 0.f, 0.f, 0.f);
}

// ---------------------------------------------------------------------------
// Kernel 2: hw = (h @ W) * norm   via V_WMMA_F32_16X16X4_F32.
// Block = 192 threads = 6 waves = one M-tile (16 rows); wave w owns N-tile w.
// Exact-fit grid (3125 blocks): EXEC is all-ones for every WMMA.
// Weight staged through LDS in 64x96 chunks (24 KB).
// ---------------------------------------------------------------------------
__global__ __launch_bounds__(192) void gcn_gemm_norm_kernel(
    const float* __restrict__ h,     // [N_NODES, IN_F]
    const float* __restrict__ w,     // [IN_F, OUT_F]
    const float* __restrict__ norm,  // [N_NODES]
    float* __restrict__ hw)          // [N_NODES, OUT_F]
{
    __shared__ float wlds[64 * OUT_F];  // 24 KB K-chunk of the weight

    const int lane  = threadIdx.x & 31;
    const int ntile = threadIdx.x >> 5;       // 0..5
    const int mtile = blockIdx.x;             // 0..3124

    // A-fragment addressing (16x4 f32): lane m = L&15, K-pair = 2*(L>>4)
    const int m     = lane & 15;
    const int khalf = (lane >> 4) << 1;       // 0 or 2
    const int row   = mtile * 16 + m;
    const float* arow = h + (size_t)row * IN_F + khalf;

    // B-fragment / C addressing: N = L&15 within this wave's N-tile
    const int ncol = ntile * 16 + (lane & 15);

    v8f c = {0.f, 0.f, 0.f, 0.f, 0.f, 0.f, 0.f, 0.f};

    for (int kc = 0; kc < IN_F; kc += 64) {
        __syncthreads();  // previous chunk fully consumed before overwrite
        // cooperative load of 64x96 weight chunk: 1536 float4 / 192 threads
        {
            const float4* wsrc = (const float4*)(w + (size_t)kc * OUT_F);
            float4*       wdst = (float4*)wlds;
            #pragma unroll
            for (int i = 0; i < 8; ++i)
                wdst[threadIdx.x + i * 192] = wsrc[threadIdx.x + i * 192];
        }
        __syncthreads();

        #pragma unroll
        for (int k = 0; k < 64; k += 4) {
            // A: two contiguous f32 per lane -> global_load_b64
            v2f a = *(const v2f*)(arow + kc + k);
            // B 4x16 f32: lane n = L&15, K-pair = khalf (from LDS)
            v2f b;
            b.x = wlds[(k + khalf)     * OUT_F + ncol];
            b.y = wlds[(k + khalf + 1) * OUT_F + ncol];
            c = __builtin_amdgcn_wmma_f32_16x16x4_f32(
                    /*neg_a=*/false, a, /*neg_b=*/false, b,
                    /*c_mod=*/(short)0, c, /*reuse_a=*/false, /*reuse_b=*/false);
        }
    }

    // C/D layout: VGPR v, lane L -> M = v + 8*(L>>4), N = L&15
    const int nhalf = lane >> 4;
    #pragma unroll
    for (int v = 0; v < 8; ++v) {
        int orow = mtile * 16 + v + 8 * nhalf;
        hw[(size_t)orow * OUT_F + ncol] = c[v] * norm[orow];
    }
}

// ---------------------------------------------------------------------------
// Kernel 3: scatter-add messages. 24 threads/edge, float4 per thread.
// agg (19.2 MB) fits in the 192 MB L2 -> atomics resolve at L2 atomic units.
// ---------------------------------------------------------------------------
__global__ void gcn_scatter_kernel(const float* __restrict__ hw,
                                   const int*   __restrict__ src,
                                   const int*   __restrict__ dst,
                                   float*       __restrict__ agg)
{
    const int CHUNKS = OUT_F / 4;  // 24
    long long t = (long long)blockIdx.x * blockDim.x + threadIdx.x;
    if (t >= (long long)N_EDGES * CHUNKS) return;
    int e = (int)(t / CHUNKS);
    int c = (int)(t % CHUNKS);
    int s = src[e];
    int d = dst[e];
    float4 v = *(const float4*)(hw + (size_t)s * OUT_F + c * 4);
    float* ap = agg + (size_t)d * OUT_F + c * 4;
    atomicAdd(ap + 0, v.x);
    atomicAdd(ap + 1, v.y);
    atomicAdd(ap + 2, v.z);
    atomicAdd(ap + 3, v.w);
}

// ---------------------------------------------------------------------------
// Kernel 4: out = relu(agg * norm), streaming float4 pass.
// ---------------------------------------------------------------------------
__global__ void gcn_finish_kernel(const float* __restrict__ agg,
                                  const float* __restrict__ norm,
                                  float*       __restrict__ out)
{
    const int n4 = N_NODES * OUT_F / 4;
    int i = blockIdx.x * blockDim.x + threadIdx.x;
    if (i >= n4) return;
    int rowq = i / (OUT_F / 4);
    float nv = norm[rowq];
    float4 v = ((const float4*)agg)[i];
    float4 o;
    o.x = fmaxf(v.x * nv, 0.f);
    o.y = fmaxf(v.y * nv, 0.f);
    o.z = fmaxf(v.z * nv, 0.f);
    o.w = fmaxf(v.w * nv, 0.f);
    ((float4*)out)[i] = o;
}

// ---------------------------------------------------------------------------
extern "C" void kernel_launch(void* const* d_in, const int* in_sizes, int n_in,
                              void* d_out, int out_size, void* d_ws, size_t ws_size,
                              hipStream_t stream) {
    const float* h    = (const float*)d_in[0];  // [50000, 256]
    const float* norm = (const float*)d_in[1];  // [50000, 1]
    const float* w    = (const float*)d_in[2];  // [256, 96]
    const int*   src  = (const int*)d_in[3];    // [800000]
    const int*   dst  = (const int*)d_in[4];    // [800000]
    float* out = (float*)d_out;

    // Workspace layout: hw [50000*96] | agg [50000*96]  (2 x 19.2 MB)
    float* hw  = (float*)d_ws;
    float* agg = hw + (size_t)N_NODES * OUT_F;

    // 1) zero agg
    const int n4 = N_NODES * OUT_F / 4;
    gcn_zero_kernel<<<(n4 + 255) / 256, 256, 0, stream>>>((float4*)agg, n4);

    // 2) hw = (h @ W) * norm  — one block per 16-row M-tile, 6 WMMA waves each
    gcn_gemm_norm_kernel<<<N_NODES / 16, 192, 0, stream>>>(h, w, norm, hw);

    // 3) scatter-add over edges
    long long nthr = (long long)N_EDGES * (OUT_F / 4);
    gcn_scatter_kernel<<<(int)((nthr + 255) / 256), 256, 0, stream>>>(hw, src, dst, agg);

    // 4) out = relu(agg * norm)
    gcn_finish_kernel<<<(n4 + 255) / 256, 256, 0, stream>>>(agg, norm, out);
}